// DiscriminativeLoss_77713138254116
// MI455X (gfx1250) — compile-verified
//
#include <hip/hip_runtime.h>
#include <hip/hip_bf16.h>

// ---------------- problem constants ----------------
#define BSZ   8
#define ND    32            // embedding dims
#define KI    24            // max instances
#define KPAD  32            // K padded for 2x 16-wide WMMA tiles
#define HW    (384*384)     // 147456 pixels per sample
#define CHUNKS 32           // pass-1 chunks per batch
#define CHUNK  (HW/CHUNKS)  // 4608 pixels per block
#define DELTA_V 0.5f
#define MARGIN  3.0f        // 2*DELTA_D
#define GAMMA   0.001f

// ---------------- workspace layout (floats unless noted) ----------------
#define OFF_CNT   64                 // counts[BSZ][KI]           (192)
#define OFF_SUM   256                // sums  [BSZ][KI][ND]       (6144)
#define OFF_MEAN  6400               // means [BSZ][KI][ND]       (6144)
#define OFF_M2    12544              // m2    [BSZ][KI]           (192)
#define OFF_PART  12800              // per-wave var partials     (NPART)
#define NPART     4096               // 512 blocks * 8 waves in pass 3
#define ZEROF     6400               // scalars + counts + sums need zeroing
#define LAB_OFF_BYTES ((size_t)(OFF_PART + NPART) * 4)  // uint8 labels after

typedef float v2f __attribute__((ext_vector_type(2)));
typedef float v8f __attribute__((ext_vector_type(8)));

// D(16x16,f32) = A(16pix x 4d, f32) * B(4d x 16k, f32) + C
__device__ __forceinline__ v8f wmma_f32_16x16x4(v2f a, v2f b, v8f c) {
  return __builtin_amdgcn_wmma_f32_16x16x4_f32(
      /*neg_a=*/false, a, /*neg_b=*/false, b,
      /*c_mod=*/(short)0, c, /*reuse_a=*/false, /*reuse_b=*/false);
}

__device__ __forceinline__ float wave_red(float v) {
  #pragma unroll
  for (int o = 16; o > 0; o >>= 1) v += __shfl_xor(v, o, 32);
  return v;
}

// ---------------- P0: zero accumulators ----------------
__global__ void k_zero(float* ws) {
  int i = blockIdx.x * 256 + threadIdx.x;
  if (i < ZEROF) ws[i] = 0.0f;
}

// ---------------- P1: labels + segmented sums/counts ----------------
// Streams target (find one-hot label) and pred (accumulate), fully coalesced.
__global__ void k_stats(const float* __restrict__ pred,
                        const float* __restrict__ tgt,
                        float* __restrict__ ws,
                        unsigned char* __restrict__ labels) {
  __shared__ float s_sum[KI][ND + 1];
  __shared__ float s_cnt[KI];
  const int tid = threadIdx.x;
  const int b = blockIdx.x / CHUNKS;
  const int l0 = (blockIdx.x % CHUNKS) * CHUNK;

  for (int i = tid; i < KI * (ND + 1); i += 256) (&s_sum[0][0])[i] = 0.0f;
  if (tid < KI) s_cnt[tid] = 0.0f;
  __syncthreads();

  const float* tb = tgt  + (size_t)b * KI * HW;
  const float* pb = pred + (size_t)b * ND * HW;

  for (int l = l0 + tid; l < l0 + CHUNK; l += 256) {
    int lab = 0;
    #pragma unroll
    for (int k = 0; k < KI; ++k) {
      float t = tb[(size_t)k * HW + l];
      lab = (t > 0.5f) ? k : lab;
    }
    labels[(size_t)b * HW + l] = (unsigned char)lab;
    #pragma unroll
    for (int d = 0; d < ND; ++d) {
      float v = pb[(size_t)d * HW + l];
      atomicAdd(&s_sum[lab][d], v);            // ds_add_f32
    }
    atomicAdd(&s_cnt[lab], 1.0f);
  }
  __syncthreads();

  float* gsum = ws + OFF_SUM + (size_t)b * KI * ND;
  float* gcnt = ws + OFF_CNT + (size_t)b * KI;
  for (int i = tid; i < KI * ND; i += 256)
    atomicAdd(&gsum[i], s_sum[i / ND][i % ND]);
  if (tid < KI) atomicAdd(&gcnt[tid], s_cnt[tid]);
}

// ---------------- P2: finalize means, WMMA mm = means*means^T, dist+reg ----
__global__ void k_meandist(const int* __restrict__ nobj, float* __restrict__ ws) {
  __shared__ float s_mean[BSZ][KPAD][ND + 1];   // padded, zero-filled
  __shared__ float s_m2[BSZ][KPAD];
  __shared__ float s_dr[BSZ][2];
  const int tid = threadIdx.x;

  for (int i = tid; i < BSZ * KPAD * (ND + 1); i += 256) (&s_mean[0][0][0])[i] = 0.0f;
  for (int i = tid; i < BSZ * KPAD; i += 256) (&s_m2[0][0])[i] = 0.0f;
  __syncthreads();

  if (tid < BSZ * KI) {                 // one thread per (b,k)
    int b = tid / KI, k = tid % KI;
    float cnt = ws[OFF_CNT + b * KI + k];
    float inv = 1.0f / fmaxf(cnt, 1.0f);
    float valid = (k < nobj[b]) ? 1.0f : 0.0f;
    float m2 = 0.0f;
    for (int d = 0; d < ND; ++d) {
      float m = ws[OFF_SUM + (b * KI + k) * ND + d] * inv * valid;
      s_mean[b][k][d] = m;
      ws[OFF_MEAN + (b * KI + k) * ND + d] = m;
      m2 += m * m;
    }
    s_m2[b][k] = m2;
    ws[OFF_M2 + b * KI + k] = m2;
  }
  __syncthreads();

  const int wave = tid >> 5, lane = tid & 31;
  const int l16 = lane & 15, half = lane >> 4;
  if (wave < BSZ) {
    const int b = wave;
    const int n = nobj[b];
    const float nf = (float)n;
    float hingeSum = 0.0f;

    #pragma unroll
    for (int ka = 0; ka < 2; ++ka)
      #pragma unroll
      for (int kb = 0; kb < 2; ++kb) {
        v8f c = {};
        #pragma unroll
        for (int dc = 0; dc < 8; ++dc) {
          const int d0 = dc * 4;
          v2f a, bm;
          a.x  = s_mean[b][ka * 16 + l16][d0 + half * 2];
          a.y  = s_mean[b][ka * 16 + l16][d0 + half * 2 + 1];
          bm.x = s_mean[b][kb * 16 + l16][d0 + half * 2];
          bm.y = s_mean[b][kb * 16 + l16][d0 + half * 2 + 1];
          c = wmma_f32_16x16x4(a, bm, c);
        }
        #pragma unroll
        for (int r = 0; r < 8; ++r) {
          int ki = ka * 16 + half * 8 + r;     // C row: M = half*8 + r
          int kj = kb * 16 + l16;              // C col: N = lane%16
          float d2 = fmaxf(s_m2[b][ki] + s_m2[b][kj] - 2.0f * c[r], 0.0f);
          float md = (d2 > 1e-12f) ? sqrtf(d2) : 0.0f;
          bool mask = (ki != kj) && (ki < n) && (kj < n);
          float h = fmaxf(MARGIN - md, 0.0f);
          hingeSum += mask ? h * h : 0.0f;
        }
      }

    float m2l = (lane < KI && lane < n) ? s_m2[b][lane] : 0.0f;
    float reg = (m2l > 1e-12f) ? sqrtf(m2l) : 0.0f;

    hingeSum = wave_red(hingeSum);
    reg      = wave_red(reg);
    if (lane == 0) {
      s_dr[b][0] = hingeSum / (nf * (nf - 1.0f));
      s_dr[b][1] = reg / nf;
    }
  }
  __syncthreads();
  if (tid == 0) {
    float ds = 0.0f, rs = 0.0f;
    for (int b = 0; b < BSZ; ++b) { ds += s_dr[b][0]; rs += s_dr[b][1]; }
    ws[1] = ds / (float)BSZ + GAMMA * (rs / (float)BSZ);
  }
}

// ---------------- P3: var term; WMMA pm = pred_tile * means^T ----------------
__global__ void k_var(const float* __restrict__ pred,
                      const unsigned char* __restrict__ labels,
                      float* __restrict__ ws) {
  __shared__ float s_mean[BSZ][KPAD][ND + 1];   // stride 33 -> conflict-free
  __shared__ float s_m2[BSZ][KPAD];
  const int tid = threadIdx.x;

  for (int i = tid; i < BSZ * KPAD * (ND + 1); i += 256) (&s_mean[0][0][0])[i] = 0.0f;
  for (int i = tid; i < BSZ * KPAD; i += 256) (&s_m2[0][0])[i] = 0.0f;
  __syncthreads();
  for (int i = tid; i < BSZ * KI * ND; i += 256)
    s_mean[i / (KI * ND)][(i / ND) % KI][i % ND] = ws[OFF_MEAN + i];
  for (int i = tid; i < BSZ * KI; i += 256)
    s_m2[i / KI][i % KI] = ws[OFF_M2 + i];
  __syncthreads();

  const int lane = tid & 31, wave = tid >> 5;
  const int l16 = lane & 15, half = lane >> 4;
  const int gw = blockIdx.x * 8 + wave;
  const int nW = gridDim.x * 8;
  const int TPB = HW / 16;                       // 16-pixel tiles per batch
  float acc = 0.0f;

  for (int t = gw; t < BSZ * TPB; t += nW) {
    const int b = t / TPB;
    const int l0 = (t % TPB) * 16;
    const float* pb = pred + (size_t)b * ND * HW + l0 + l16;
    int lab = labels[(size_t)b * HW + l0 + l16]; // all lanes load (dup ok)

    v8f c0 = {}, c1 = {};
    float p2p = 0.0f;
    #pragma unroll
    for (int dc = 0; dc < 8; ++dc) {
      const int d0 = dc * 4;
      v2f a, b0, b1;
      // A: lanes 0-15 hold d0,d0+1; lanes 16-31 hold d0+2,d0+3 (16x4 f32 layout)
      a.x = pb[(size_t)(d0 + half * 2) * HW];
      a.y = pb[(size_t)(d0 + half * 2 + 1) * HW];
      p2p += a.x * a.x + a.y * a.y;
      b0.x = s_mean[b][l16][d0 + half * 2];
      b0.y = s_mean[b][l16][d0 + half * 2 + 1];
      b1.x = s_mean[b][16 + l16][d0 + half * 2];
      b1.y = s_mean[b][16 + l16][d0 + half * 2 + 1];
      c0 = wmma_f32_16x16x4(a, b0, c0);          // k = 0..15
      c1 = wmma_f32_16x16x4(a, b1, c1);          // k = 16..31 (>=24 never matches)
    }
    float p2 = p2p + __shfl_xor(p2p, 16, 32);    // full ||p||^2, pixel = lane%16

    #pragma unroll
    for (int r = 0; r < 8; ++r) {
      const int srcpix = half * 8 + r;           // C row -> pixel in tile
      float p2s = __shfl(p2, srcpix, 32);
      int  labs = __shfl(lab, srcpix, 32);
      {
        float d2 = fmaxf(p2s + s_m2[b][l16] - 2.0f * c0[r], 0.0f);
        float dd = (d2 > 1e-12f) ? sqrtf(d2) : 0.0f;
        float h = fmaxf(dd - DELTA_V, 0.0f);
        if (labs == l16) acc += h * h;
      }
      {
        float d2 = fmaxf(p2s + s_m2[b][16 + l16] - 2.0f * c1[r], 0.0f);
        float dd = (d2 > 1e-12f) ? sqrtf(d2) : 0.0f;
        float h = fmaxf(dd - DELTA_V, 0.0f);
        if (labs == 16 + l16) acc += h * h;
      }
    }
  }
  acc = wave_red(acc);
  if (lane == 0) ws[OFF_PART + gw] = acc;        // deterministic partials
}

// ---------------- P4: final reduce + combine ----------------
__global__ void k_final(const float* __restrict__ ws, float* __restrict__ out) {
  __shared__ float red[256];
  float s = 0.0f;
  for (int i = threadIdx.x; i < NPART; i += 256) s += ws[OFF_PART + i];
  red[threadIdx.x] = s;
  __syncthreads();
  for (int st = 128; st > 0; st >>= 1) {
    if (threadIdx.x < st) red[threadIdx.x] += red[threadIdx.x + st];
    __syncthreads();
  }
  if (threadIdx.x == 0)
    out[0] = red[0] / (float)((size_t)BSZ * HW) + ws[1];
}

extern "C" void kernel_launch(void* const* d_in, const int* in_sizes, int n_in,
                              void* d_out, int out_size, void* d_ws, size_t ws_size,
                              hipStream_t stream) {
  (void)in_sizes; (void)n_in; (void)out_size; (void)ws_size;
  const float* pred = (const float*)d_in[0];
  const float* tgt  = (const float*)d_in[1];
  const int*   nobj = (const int*)d_in[2];
  float* ws = (float*)d_ws;
  unsigned char* labels = (unsigned char*)d_ws + LAB_OFF_BYTES;
  float* out = (float*)d_out;

  k_zero    <<<(ZEROF + 255) / 256, 256, 0, stream>>>(ws);
  k_stats   <<<BSZ * CHUNKS,        256, 0, stream>>>(pred, tgt, ws, labels);
  k_meandist<<<1,                   256, 0, stream>>>(nobj, ws);
  k_var     <<<512,                 256, 0, stream>>>(pred, labels, ws);
  k_final   <<<1,                   256, 0, stream>>>(ws, out);
}